// VectorQuantizer_63385127355197
// MI455X (gfx1250) — compile-verified
//
#include <hip/hip_runtime.h>

// CDNA5 / gfx1250 vector-quantizer:
//   out[i] = argmin_k ( ||emb_k||^2 - 2 * sum_j emb[k][j] * x[i]^(j+1) )  (as float)
//
// Matrix-pipe mapping (V_WMMA_F32_16X16X4_F32, K extended to 24 = 6 chunks):
//   A (16x24): rows 0-4  = -2*emb for GROUP0 in K-chunks 0-2, zero elsewhere
//              rows 8-12 = -2*emb for GROUP1 in K-chunks 3-5, zero elsewhere
//   B (24x16): cols n carry powers x^1..x^10 of group0 element n (chunks 0-2)
//              and of group1 element n (chunks 3-5)
//   C        : ||emb_k||^2 broadcast into accumulator VGPRs 0..4 (all lanes)
// D layout (ISA 7.12.2): VGPR v = row v in lanes 0-15 AND row v+8 in lanes 16-31,
// so after 6 chained WMMAs each lane L holds the 5 scores of element base+L in
// acc[0..4]: low lanes via rows 0-4 (group0), high lanes via rows 8-12 (group1).
// => one per-lane argmin, full-wave store, no output shuffle.
// EXEC is all-ones at every WMMA (no predication in the main loop).

typedef __attribute__((ext_vector_type(2))) float v2f;
typedef __attribute__((ext_vector_type(8))) float v8f;

#define NUM_EMB 5
#define EMB_DIM 10
#define SWAPX16 0x401F   // group-of-32 ds_swizzle: xor=0x10, or=0, and=0x1f

__device__ __forceinline__ float swap_halves(float v) {
  return __int_as_float(__builtin_amdgcn_ds_swizzle(__float_as_int(v), SWAPX16));
}

__device__ __forceinline__ float argmin5(const v8f& acc) {
  float best = acc[0];
  float bi   = 0.0f;
#pragma unroll
  for (int k = 1; k < NUM_EMB; ++k) {
    bool lt = acc[k] < best;          // first-min tie-break, matches jnp.argmin
    best = lt ? acc[k] : best;
    bi   = lt ? (float)k : bi;
  }
  return bi;
}

__global__ void __launch_bounds__(256)
vq_wmma_kernel(const float* __restrict__ X,
               const float* __restrict__ E,
               float* __restrict__ out,
               int n) {
  const int lane = threadIdx.x & 31;
  const bool hi  = lane >= 16;
  const int  m   = lane & 15;      // A-matrix row owned by this lane
  const int  k0  = hi ? 2 : 0;     // A layout: lanes 16-31 carry K=2,3 of each chunk

  // ---- A operands ----
  // rows 0..4  : -2*emb[r][kk]   (kk = K within chunks 0-2, i.e. global K 0..11)
  auto coefLo = [&](int kk) -> float {
    return (m < NUM_EMB && kk < EMB_DIM) ? -2.0f * E[m * EMB_DIM + kk] : 0.0f;
  };
  // rows 8..12 : -2*emb[r-8][kk] (kk = K within chunks 3-5, i.e. global K 12..23)
  auto coefHi = [&](int kk) -> float {
    return (m >= 8 && m < 8 + NUM_EMB && kk < EMB_DIM)
               ? -2.0f * E[(m - 8) * EMB_DIM + kk] : 0.0f;
  };
  const v2f A0 = { coefLo(0 + k0), coefLo(1 + k0) };   // K  0..3
  const v2f A1 = { coefLo(4 + k0), coefLo(5 + k0) };   // K  4..7
  const v2f A2 = { coefLo(8 + k0), coefLo(9 + k0) };   // K  8..11 (cols 10,11 pad)
  const v2f A3 = { coefHi(0 + k0), coefHi(1 + k0) };   // K 12..15
  const v2f A4 = { coefHi(4 + k0), coefHi(5 + k0) };   // K 16..19
  const v2f A5 = { coefHi(8 + k0), coefHi(9 + k0) };   // K 20..23 (pad cols)

  // ---- C init: ||emb_v||^2 in acc VGPR v, ALL lanes (covers rows v and v+8) ----
  v8f cinit = {};
#pragma unroll
  for (int v = 0; v < NUM_EMB; ++v) {
    float s = 0.0f;
#pragma unroll
    for (int i = 0; i < EMB_DIM; ++i) { float e = E[v * EMB_DIM + i]; s += e * e; }
    cinit[v] = s;
  }

  // B chunk builder: xg holds one element's value in both half-waves;
  // low lanes supply odd powers (rows 4c,4c+2), high lanes even (rows 4c+1,4c+3).
  auto ladder = [&](float xg, v2f& B0, v2f& B1, v2f& B2) {
    float x2 = xg * xg;
    float b0 = hi ? x2 : xg;                 // x^1 / x^2
    float b1 = b0 * x2;                      // x^3 / x^4
    float b2 = b1 * x2;                      // x^5 / x^6
    float b3 = b2 * x2;                      // x^7 / x^8
    float b4 = b3 * x2;                      // x^9 / x^10
    B0 = { b0, b1 };
    B1 = { b2, b3 };
    B2 = { b4, 0.0f };                       // pad rows
  };

  const int waves_per_block = blockDim.x >> 5;
  const int wave   = blockIdx.x * waves_per_block + (threadIdx.x >> 5);
  const int nwaves = gridDim.x * waves_per_block;
  const int npairs = n >> 5;                 // full 32-element iterations

  // ---- main loop: 32 elements/iter, full-wave VMEM, software-pipelined load ----
  if (wave < npairs) {                       // wave-uniform guard
    float xv = X[(wave << 5) + lane];
    for (int p = wave; p < npairs; p += nwaves) {
      const int base  = p << 5;
      const int nextp = p + nwaves;
      // preload next iteration (clamped, wave-uniform select -> no predication)
      const int naddr = (nextp < npairs ? (nextp << 5) : base) + lane;
      const float xnext = X[naddr];

      const float sw = swap_halves(xv);      // exchange half-waves (LDS permute)
      const float g0 = hi ? sw : xv;         // elements 0..15 in both halves
      const float g1 = hi ? xv : sw;         // elements 16..31 in both halves

      v2f B0, B1, B2;
      v8f acc = cinit;
      ladder(g0, B0, B1, B2);
      acc = __builtin_amdgcn_wmma_f32_16x16x4_f32(false, A0, false, B0, (short)0, acc, false, false);
      acc = __builtin_amdgcn_wmma_f32_16x16x4_f32(false, A1, false, B1, (short)0, acc, false, false);
      acc = __builtin_amdgcn_wmma_f32_16x16x4_f32(false, A2, false, B2, (short)0, acc, false, false);
      ladder(g1, B0, B1, B2);
      acc = __builtin_amdgcn_wmma_f32_16x16x4_f32(false, A3, false, B0, (short)0, acc, false, false);
      acc = __builtin_amdgcn_wmma_f32_16x16x4_f32(false, A4, false, B1, (short)0, acc, false, false);
      acc = __builtin_amdgcn_wmma_f32_16x16x4_f32(false, A5, false, B2, (short)0, acc, false, false);

      out[base + lane] = argmin5(acc);       // full-wave store of 32 results
      xv = xnext;
    }
  }

  // ---- tail (n % 32): wave 0 only (wave-uniform branch -> EXEC full at WMMA) ----
  if (wave == 0) {
    for (int tb = npairs << 5; tb < n; tb += 16) {
      const int idx = tb + m;
      float x = (idx < n) ? X[idx] : 0.0f;   // element value in both halves
      v2f B0, B1, B2;
      v8f acc = cinit;
      ladder(x, B0, B1, B2);
      acc = __builtin_amdgcn_wmma_f32_16x16x4_f32(false, A0, false, B0, (short)0, acc, false, false);
      acc = __builtin_amdgcn_wmma_f32_16x16x4_f32(false, A1, false, B1, (short)0, acc, false, false);
      acc = __builtin_amdgcn_wmma_f32_16x16x4_f32(false, A2, false, B2, (short)0, acc, false, false);
      const float bi = argmin5(acc);         // valid in lanes 0..15
      if (!hi && idx < n) out[idx] = bi;
    }
  }
}

extern "C" void kernel_launch(void* const* d_in, const int* in_sizes, int n_in,
                              void* d_out, int out_size, void* d_ws, size_t ws_size,
                              hipStream_t stream) {
  const float* x   = (const float*)d_in[0];
  const float* emb = (const float*)d_in[1];
  float* out       = (float*)d_out;
  const int n      = in_sizes[0];            // 4096*2048 = 8,388,608

  const int threads = 256;                   // 8 wave32 per block
  const int npairs  = (n + 31) >> 5;
  int blocks = (npairs + 7) / 8;             // one 32-elem pair per wave if small
  if (blocks > 2048) blocks = 2048;          // ~16K waves, ~16 pairs/wave
  if (blocks < 1) blocks = 1;

  vq_wmma_kernel<<<blocks, threads, 0, stream>>>(x, emb, out, n);
}